// HANLayer_3822520893712
// MI455X (gfx1250) — compile-verified
//
#include <hip/hip_runtime.h>
#include <math.h>

#define NN 50000
#define EE 800000
#define MM 3
#define IN_SZ 128
#define HH 8
#define OO 32
#define DD 256     // H*O
#define HID 128

typedef __attribute__((ext_vector_type(2))) float v2f;
typedef __attribute__((ext_vector_type(8))) float v8f;

// ---------------------------------------------------------------------------
// feat = h[16-row strip] @ W  via V_WMMA_F32_16X16X4_F32. One wave per strip.
// A frag layout (32-bit A 16x4): lane<16 holds K=k,k+1 ; lane>=16 holds K=k+2,k+3 ; M=lane&15
// B frag layout (32-bit B 4x16): lane<16 holds rows k,k+1 ; lane>=16 rows k+2,k+3 ; N=lane&15
// C/D: vgpr r -> M = r + 8*(lane>=16), N = lane&15
// ---------------------------------------------------------------------------
__global__ __launch_bounds__(32) void feat_gemm_kernel(
    const float* __restrict__ h, const float* __restrict__ W,
    float* __restrict__ feat) {
  const int lane = threadIdx.x;
  const int half = lane >> 4;       // 0 or 1
  const int l16  = lane & 15;
  const int row0 = blockIdx.x * 16; // N = 3125*16 exactly

  // Preload A strip: 16 rows x 128 K -> 32 K-steps, 2 floats/lane/step
  v2f a[32];
  const float* arow = h + (size_t)(row0 + l16) * IN_SZ;
#pragma unroll
  for (int kk = 0; kk < 32; ++kk) {
    int k = kk * 4 + half * 2;
    a[kk].x = arow[k];
    a[kk].y = arow[k + 1];
  }

  for (int ct = 0; ct < 16; ++ct) {        // 16 col tiles of 16 -> D=256
    v8f acc = {};
#pragma unroll
    for (int kk = 0; kk < 32; ++kk) {
      int k = kk * 4 + half * 2;
      v2f b;
      b.x = W[(size_t)k * DD + ct * 16 + l16];
      b.y = W[(size_t)(k + 1) * DD + ct * 16 + l16];
      acc = __builtin_amdgcn_wmma_f32_16x16x4_f32(
          false, a[kk], false, b, (short)0, acc, false, false);
    }
#pragma unroll
    for (int r = 0; r < 8; ++r) {
      feat[(size_t)(row0 + half * 8 + r) * DD + ct * 16 + l16] = acc[r];
    }
  }
}

// el/er: per (node, head) 32-length dots
__global__ void eler_kernel(const float* __restrict__ feat,
                            const float* __restrict__ al,
                            const float* __restrict__ ar,
                            float* __restrict__ el, float* __restrict__ er) {
  int i = blockIdx.x * blockDim.x + threadIdx.x;
  if (i >= NN * HH) return;
  int n = i >> 3, hh = i & 7;
  const float* f  = feat + (size_t)n * DD + hh * OO;
  const float* a1 = al + hh * OO;
  const float* a2 = ar + hh * OO;
  float s1 = 0.f, s2 = 0.f;
#pragma unroll
  for (int o = 0; o < OO; ++o) { float fv = f[o]; s1 += fv * a1[o]; s2 += fv * a2[o]; }
  el[i] = s1; er[i] = s2;
}

__global__ void minit_kernel(float* __restrict__ m, int count) {
  int i = blockIdx.x * blockDim.x + threadIdx.x;
  if (i < count) m[i] = -INFINITY;
}

__device__ __forceinline__ void atomicMaxFloat(float* addr, float val) {
  if (val >= 0.f) atomicMax((int*)addr, __float_as_int(val));
  else            atomicMin((unsigned int*)addr, __float_as_uint(val));
}

__global__ void edge_max_kernel(const int* __restrict__ src,
                                const int* __restrict__ dst,
                                const float* __restrict__ el,
                                const float* __restrict__ er,
                                float* __restrict__ mbuf) {
  int e = blockIdx.x * blockDim.x + threadIdx.x;
  if (e >= EE) return;
  int s = src[e], d = dst[e];
#pragma unroll
  for (int hh = 0; hh < HH; ++hh) {
    float v = el[s * HH + hh] + er[d * HH + hh];
    v = v > 0.f ? v : 0.2f * v;
    atomicMaxFloat(&mbuf[d * HH + hh], v);
  }
}

__global__ void edge_sum_kernel(const int* __restrict__ src,
                                const int* __restrict__ dst,
                                const float* __restrict__ el,
                                const float* __restrict__ er,
                                const float* __restrict__ mbuf,
                                float* __restrict__ den) {
  int e = blockIdx.x * blockDim.x + threadIdx.x;
  if (e >= EE) return;
  int s = src[e], d = dst[e];
#pragma unroll
  for (int hh = 0; hh < HH; ++hh) {
    float v = el[s * HH + hh] + er[d * HH + hh];
    v = v > 0.f ? v : 0.2f * v;
    atomicAdd(&den[d * HH + hh], expf(v - mbuf[d * HH + hh]));
  }
}

// One 256-thread block per edge: 8 threads compute per-head alpha into LDS,
// then all 256 threads do coalesced gather(feat[src]) + atomicAdd(z[dst]).
// zm points at z + m*DD; node stride is MM*DD.
__global__ __launch_bounds__(256) void edge_scatter_kernel(
    const int* __restrict__ src, const int* __restrict__ dst,
    const float* __restrict__ el, const float* __restrict__ er,
    const float* __restrict__ mbuf, const float* __restrict__ den,
    const float* __restrict__ feat, float* __restrict__ zm) {
  __shared__ float alpha_s[HH];
  int e = blockIdx.x;
  int s = src[e], d = dst[e];
  if (threadIdx.x < HH) {
    int hh = threadIdx.x;
    float v = el[s * HH + hh] + er[d * HH + hh];
    v = v > 0.f ? v : 0.2f * v;
    float dn = den[d * HH + hh];
    dn = (dn == 0.f) ? 1.f : dn;
    alpha_s[hh] = expf(v - mbuf[d * HH + hh]) / dn;
  }
  __syncthreads();
  int dd = threadIdx.x;  // 0..255
  atomicAdd(&zm[(size_t)d * (MM * DD) + dd],
            alpha_s[dd >> 5] * feat[(size_t)s * DD + dd]);
}

__global__ void bias_elu_kernel(float* __restrict__ zm,
                                const float* __restrict__ bias) {
  int i = blockIdx.x * blockDim.x + threadIdx.x;
  if (i >= NN * DD) return;
  int n = i >> 8, dd = i & 255;
  size_t idx = (size_t)n * (MM * DD) + dd;
  float x = zm[idx] + bias[dd];
  zm[idx] = x > 0.f ? x : expm1f(x);
}

// Semantic attention score: block per (n,m) row; accumulate only the per-m sums.
__global__ __launch_bounds__(128) void semantic_kernel(
    const float* __restrict__ z, const float* __restrict__ W1,
    const float* __restrict__ b1, const float* __restrict__ W2,
    float* __restrict__ wsum) {
  __shared__ float zrow[DD];
  __shared__ float red[HID];
  int nm = blockIdx.x;  // n*MM + m
  int t = threadIdx.x;  // 0..127
  zrow[t]       = z[(size_t)nm * DD + t];
  zrow[t + 128] = z[(size_t)nm * DD + 128 + t];
  __syncthreads();
  float s = b1[t];
#pragma unroll 8
  for (int i = 0; i < DD; ++i) s += zrow[i] * W1[(size_t)i * HID + t];
  red[t] = tanhf(s) * W2[t];
  __syncthreads();
  for (int off = 64; off > 0; off >>= 1) {
    if (t < off) red[t] += red[t + off];
    __syncthreads();
  }
  if (t == 0) atomicAdd(&wsum[nm % MM], red[0]);
}

__global__ void beta_kernel(const float* __restrict__ wsum,
                            float* __restrict__ beta) {
  float w0 = wsum[0] * (1.f / NN);
  float w1 = wsum[1] * (1.f / NN);
  float w2 = wsum[2] * (1.f / NN);
  float mx = fmaxf(w0, fmaxf(w1, w2));
  float e0 = expf(w0 - mx), e1 = expf(w1 - mx), e2 = expf(w2 - mx);
  float inv = 1.f / (e0 + e1 + e2);
  beta[0] = e0 * inv; beta[1] = e1 * inv; beta[2] = e2 * inv;
}

__global__ void final_kernel(const float* __restrict__ z,
                             const float* __restrict__ beta,
                             float* __restrict__ out) {
  int i = blockIdx.x * blockDim.x + threadIdx.x;
  if (i >= NN * DD) return;
  int n = i >> 8, dd = i & 255;
  size_t base = (size_t)n * (MM * DD) + dd;
  out[i] = beta[0] * z[base] + beta[1] * z[base + DD] + beta[2] * z[base + 2 * DD];
}

extern "C" void kernel_launch(void* const* d_in, const int* in_sizes, int n_in,
                              void* d_out, int out_size, void* d_ws, size_t ws_size,
                              hipStream_t stream) {
  const float* h       = (const float*)d_in[0];  // [M,N,IN]
  const int*   eidx    = (const int*)  d_in[1];  // [M,2,E]
  const float* W       = (const float*)d_in[2];  // [M,IN,D]
  const float* attn_l  = (const float*)d_in[3];  // [M,H,O]
  const float* attn_r  = (const float*)d_in[4];  // [M,H,O]
  const float* bias    = (const float*)d_in[5];  // [M,D]
  const float* sem_W1  = (const float*)d_in[6];  // [D,HID]
  const float* sem_b1  = (const float*)d_in[7];  // [HID]
  const float* sem_W2  = (const float*)d_in[8];  // [HID,1]
  float* out = (float*)d_out;

  float* ws   = (float*)d_ws;
  float* z    = ws;                                  // N*M*D
  float* feat = z + (size_t)NN * MM * DD;            // N*D (reused per metapath)
  float* el   = feat + (size_t)NN * DD;              // N*H
  float* er   = el + NN * HH;                        // N*H
  float* mbuf = er + NN * HH;                        // N*H
  float* den  = mbuf + NN * HH;                      // N*H
  float* wsum = den + NN * HH;                       // M
  float* beta = wsum + MM;                           // M

  hipMemsetAsync(z, 0, sizeof(float) * (size_t)NN * MM * DD, stream);
  hipMemsetAsync(wsum, 0, sizeof(float) * MM, stream);

  for (int m = 0; m < MM; ++m) {
    const float* hm   = h + (size_t)m * NN * IN_SZ;
    const float* Wm   = W + (size_t)m * IN_SZ * DD;
    const float* alm  = attn_l + (size_t)m * HH * OO;
    const float* arm  = attn_r + (size_t)m * HH * OO;
    const float* bm   = bias + (size_t)m * DD;
    const int*   srcm = eidx + (size_t)m * 2 * EE;
    const int*   dstm = srcm + EE;
    float* zm = z + (size_t)m * DD;

    feat_gemm_kernel<<<NN / 16, 32, 0, stream>>>(hm, Wm, feat);
    eler_kernel<<<(NN * HH + 255) / 256, 256, 0, stream>>>(feat, alm, arm, el, er);
    minit_kernel<<<(NN * HH + 255) / 256, 256, 0, stream>>>(mbuf, NN * HH);
    hipMemsetAsync(den, 0, sizeof(float) * NN * HH, stream);
    edge_max_kernel<<<(EE + 255) / 256, 256, 0, stream>>>(srcm, dstm, el, er, mbuf);
    edge_sum_kernel<<<(EE + 255) / 256, 256, 0, stream>>>(srcm, dstm, el, er, mbuf, den);
    edge_scatter_kernel<<<EE, 256, 0, stream>>>(srcm, dstm, el, er, mbuf, den, feat, zm);
    bias_elu_kernel<<<(NN * DD + 255) / 256, 256, 0, stream>>>(zm, bm);
  }

  semantic_kernel<<<NN * MM, 128, 0, stream>>>(z, sem_W1, sem_b1, sem_W2, wsum);
  beta_kernel<<<1, 1, 0, stream>>>(wsum, beta);
  final_kernel<<<(NN * DD + 255) / 256, 256, 0, stream>>>(z, beta, out);
}